// SpatialAttention_80126909874768
// MI455X (gfx1250) — compile-verified
//
#include <hip/hip_runtime.h>
#include <hip/hip_bf16.h>
#include <math.h>

// ---------------- problem constants ----------------
#define Ns 32
#define Cs 256
#define Ts 128
#define Vs 25
#define VP 32           // padded joint dim for WMMA tiles
#define DKs 64
#define DVs 256
#define NHs 8
#define DKH 8
#define DVH 32
#define QKVO (3*DKs + DVs)   // 448
#define EPSc 1e-5f
#define QSCALE 0.35355339059327379f  // 8^-0.5

// ---------------- workspace layout (floats) ----------------
#define OFF_DBN 0ull
#define OFF_QKV 12800ull
#define OFF_PQK (OFF_QKV + 58720256ull)
#define OFF_ATT (OFF_PQK + 16777216ull)
#define OFF_BN2 (OFF_ATT + 33554432ull)

typedef __attribute__((ext_vector_type(2))) float v2f;
typedef __attribute__((ext_vector_type(8))) float v8f;

// fp32 WMMA: D(16x16) = A(16x4) * B(4x16) + C
__device__ __forceinline__ v8f wmma4(v2f a, v2f b, v8f c) {
  return __builtin_amdgcn_wmma_f32_16x16x4_f32(false, a, false, b, (short)0, c,
                                               false, false);
}

// A-operand pattern: A[M][K] row-major.
// lane<16: rows m0+lane, K=k0+0,1 ; lane>=16: rows m0+lane-16, K=k0+2,3
__device__ __forceinline__ v2f ld_rowk(const float* m, int pitch, int m0,
                                       int k0, int lane) {
  int half = lane >> 4;
  int r = m0 + (lane & 15);
  int kk = k0 + half * 2;
  v2f o;
  o.x = m[r * pitch + kk];
  o.y = m[r * pitch + kk + 1];
  return o;
}

// B-operand pattern: B[K][N] row-major.
// lane<16: N=n0+lane, K=k0+0,1 ; lane>=16: N=n0+lane-16, K=k0+2,3
__device__ __forceinline__ v2f ld_kcol(const float* m, int pitch, int k0,
                                       int n0, int lane) {
  int half = lane >> 4;
  int c = n0 + (lane & 15);
  int kk = k0 + half * 2;
  v2f o;
  o.x = m[kk * pitch + c];
  o.y = m[(kk + 1) * pitch + c];
  return o;
}

// async global->LDS copy, 16 bytes per lane (ASYNCcnt-tracked, no VGPR data)
__device__ __forceinline__ void async_g2l_b128(unsigned lds_byte_off,
                                               const float* sbase,
                                               unsigned byte_off) {
  asm volatile("global_load_async_to_lds_b128 %0, %1, %2"
               :
               : "v"(lds_byte_off), "v"(byte_off), "s"(sbase)
               : "memory");
}
__device__ __forceinline__ void wait_async0() {
  asm volatile("s_wait_asynccnt 0x0" ::: "memory");
}

// spatial positional encoding pe[c][v]
__device__ __forceinline__ float spatial_pe(int c, int v) {
  float i2 = (float)((c >> 1) * 2);
  float d = __expf(i2 * (-9.210340371976184f / 256.0f));
  float ang = (float)v * d;
  return (c & 1) ? __cosf(ang) : __sinf(ang);
}

// ---------------- Kernel A: data_bn stats -> fused scale/shift ----------------
__global__ void kA_dbn_stats(const float* __restrict__ x,
                             const float* __restrict__ gamma,
                             const float* __restrict__ beta,
                             float* __restrict__ dbn) {
  __shared__ float s_sum[256];
  __shared__ float s_sq[256];
  int f = blockIdx.x;
  int c = f / Vs, v = f % Vs;
  int tid = threadIdx.x;
  float sum = 0.f, sq = 0.f;
  for (int i = tid; i < Ns * Ts; i += 256) {
    int n = i >> 7, t = i & 127;
    float val = x[((n * Cs + c) * Ts + t) * Vs + v];
    sum += val;
    sq += val * val;
  }
  s_sum[tid] = sum;
  s_sq[tid] = sq;
  __syncthreads();
  for (int s = 128; s > 0; s >>= 1) {
    if (tid < s) {
      s_sum[tid] += s_sum[tid + s];
      s_sq[tid] += s_sq[tid + s];
    }
    __syncthreads();
  }
  if (tid == 0) {
    float mu = s_sum[0] * (1.0f / 4096.0f);
    float var = s_sq[0] * (1.0f / 4096.0f) - mu * mu;
    float sc = rsqrtf(var + EPSc) * gamma[f];
    dbn[f] = sc;
    dbn[6400 + f] = beta[f] - mu * sc;
  }
}

// ---------------- Kernel B: QKV projection (WMMA) ----------------
// block per (n,t): stage xn and pos in LDS [256][32]; each row-tile job
// computes BOTH 16-wide N tiles sharing one A load per k-step.
__global__ void kB_qkv(const float* __restrict__ x,
                       const float* __restrict__ W,
                       const float* __restrict__ bias,
                       const float* __restrict__ dbn,
                       float* __restrict__ qkvbuf,
                       float* __restrict__ pqkbuf) {
  extern __shared__ float lds[];
  float* xn = lds;                 // 256*32
  float* pos = lds + 256 * VP;     // 256*32
  int nt = blockIdx.x;
  int n = nt / Ts, t = nt % Ts;
  int tid = threadIdx.x;

  for (int idx = tid; idx < Cs * VP; idx += 256) {
    int c = idx >> 5, v = idx & 31;
    float a = 0.f, p = 0.f;
    if (v < Vs) {
      float xv = x[((n * Cs + c) * Ts + t) * Vs + v];
      int f = c * Vs + v;
      a = xv * dbn[f] + dbn[6400 + f];
      p = xv + spatial_pe(c, v);
    }
    xn[idx] = a;
    pos[idx] = p;
  }
  __syncthreads();

  int wave = tid >> 5, lane = tid & 31;
  float* qdst = qkvbuf + (size_t)nt * QKVO * VP;
  float* pdst = pqkbuf + (size_t)nt * 128 * VP;
  int half = lane >> 4;
  int cbase = lane & 15;

  for (int rt = wave; rt < 36; rt += 8) {  // 28 xn row-tiles + 8 pos row-tiles
    const float* src;
    float* dst;
    int m0;
    if (rt < 28) { src = xn;  m0 = rt * 16;        dst = qdst; }
    else         { src = pos; m0 = (rt - 28) * 16; dst = pdst; }

    __builtin_prefetch(&W[(m0 + cbase) * Cs], 0, 1);  // branchless, once/job

    v8f acc0 = {0.f, 0.f, 0.f, 0.f, 0.f, 0.f, 0.f, 0.f};
    v8f acc1 = {0.f, 0.f, 0.f, 0.f, 0.f, 0.f, 0.f, 0.f};
    for (int k0 = 0; k0 < Cs; k0 += 4) {
      v2f a = ld_rowk(W, Cs, m0, k0, lane);
      v2f b0 = ld_kcol(src, VP, k0, 0, lane);
      v2f b1 = ld_kcol(src, VP, k0, 16, lane);
      acc0 = wmma4(a, b0, acc0);   // shared A across both N tiles
      acc1 = wmma4(a, b1, acc1);
    }
#pragma unroll
    for (int r = 0; r < 8; ++r) {
      int o = m0 + half * 8 + r;
      float s = (o < DKs) ? QSCALE : 1.0f;   // q / pq scaling
      float v0 = (acc0[r] + bias[o]) * s;
      float v1 = (acc1[r] + bias[o]) * s;
      int col0 = cbase;        // n0 = 0 tile
      int col1 = 16 + cbase;   // n0 = 16 tile
      dst[o * VP + col0] = v0;                        // col0 < 16 < Vs always
      dst[o * VP + col1] = (col1 < Vs) ? v1 : 0.f;    // zero pad columns
    }
  }
}

// ---------------- Kernel C: per-head attention (WMMA + tanh) ----------------
#define HEADSZ 3336
__global__ void kC_attn(const float* __restrict__ qkvbuf,
                        const float* __restrict__ pqkbuf,
                        float* __restrict__ attnbuf) {
  extern __shared__ float lds[];
  int nt = blockIdx.x;
  int tid = threadIdx.x;
  int wave = tid >> 5, lane = tid & 31;
  int h = wave;

  float* base = lds + wave * HEADSZ;
  float* q  = base;          // 8x32
  float* kk = base + 256;    // 8x32
  float* g  = base + 512;    // 8x32
  float* pq = base + 768;    // 8x32
  float* pk = base + 1024;   // 8x32
  float* vm = base + 1280;   // 32x32
  float* wT = base + 2304;   // 32x32 weights transposed [w][v]
  float* mu = base + 3328;   // 8

  const float* qs = qkvbuf + (size_t)nt * QKVO * VP;
  const float* ps = pqkbuf + (size_t)nt * 128 * VP;

  for (int i = lane; i < 256; i += 32) {
    q[i]  = qs[(h * 8) * VP + i];
    kk[i] = qs[(DKs + h * 8) * VP + i];
    g[i]  = qs[(2 * DKs + h * 8) * VP + i];
    pq[i] = ps[(h * 8) * VP + i];
    pk[i] = ps[(DKs + h * 8) * VP + i];
  }
  for (int i = lane; i < 1024; i += 32)
    vm[i] = qs[(3 * DKs + h * DVH) * VP + i];
  __syncthreads();

  if (lane < 8) {
    float s = 0.f;
    for (int v = 0; v < Vs; ++v) s += q[lane * VP + v];
    mu[lane] = s * (1.0f / (float)Vs);
  }
  __syncthreads();
  for (int i = lane; i < 256; i += 32) q[i] -= mu[i >> 5];   // qc
  __syncthreads();

#pragma unroll
  for (int tile = 0; tile < 4; ++tile) {
    int m0 = (tile >> 1) * 16, n0 = (tile & 1) * 16;
    v8f lp = {0.f, 0.f, 0.f, 0.f, 0.f, 0.f, 0.f, 0.f};
    v8f pr = {0.f, 0.f, 0.f, 0.f, 0.f, 0.f, 0.f, 0.f};
#pragma unroll
    for (int k0 = 0; k0 < 8; k0 += 4) {
      lp = wmma4(ld_kcol(pq, VP, k0, m0, lane), ld_kcol(pk, VP, k0, n0, lane), lp);
      pr = wmma4(ld_kcol(q,  VP, k0, m0, lane), ld_kcol(kk, VP, k0, n0, lane), pr);
    }
    int half = lane >> 4;
    int wcol = n0 + (lane & 15);
    float u = 0.f;
#pragma unroll
    for (int d = 0; d < 8; ++d) u += mu[d] * g[d * VP + wcol];
    bool wok = wcol < Vs;
#pragma unroll
    for (int r = 0; r < 8; ++r) {
      int vrow = m0 + half * 8 + r;
      float l = lp[r];
      float wgt = tanhf(pr[r] + l) + tanhf(u + l);
      wT[wcol * VP + vrow] = wok ? wgt : 0.f;  // zero pad (K rows of next GEMM)
    }
  }
  __syncthreads();

  float* outp = attnbuf + (size_t)nt * DVs * VP;
#pragma unroll
  for (int tile = 0; tile < 4; ++tile) {
    int m0 = (tile >> 1) * 16, n0 = (tile & 1) * 16;
    v8f acc = {0.f, 0.f, 0.f, 0.f, 0.f, 0.f, 0.f, 0.f};
#pragma unroll
    for (int k0 = 0; k0 < 32; k0 += 4)
      acc = wmma4(ld_rowk(vm, VP, m0, k0, lane), ld_kcol(wT, VP, k0, n0, lane), acc);
    int half = lane >> 4;
    int col = n0 + (lane & 15);
    bool ok = col < Vs;
#pragma unroll
    for (int r = 0; r < 8; ++r) {
      int e = m0 + half * 8 + r;
      outp[(h * DVH + e) * VP + col] = ok ? acc[r] : 0.f;
    }
  }
}

// ---------------- Kernel D: output projection + residual (WMMA) ----------------
// Stages attn slice via GLOBAL_LOAD_ASYNC_TO_LDS_B128 (ASYNCcnt path).
__global__ void kD_proj(const float* __restrict__ x,
                        const float* __restrict__ attnbuf,
                        const float* __restrict__ Wo,
                        const float* __restrict__ bo,
                        float* __restrict__ ybuf) {
  extern __shared__ float lds[];
  float* a_lds = lds;  // 256*32 floats, LDS offset 0
  int nt = blockIdx.x;
  int n = nt / Ts, t = nt % Ts;
  int tid = threadIdx.x;

  const float* src = attnbuf + (size_t)nt * DVs * VP;
  // 8192 floats = 2048 x b128; 8 async copies per thread, LDS dst = same offset
  for (int i = tid; i < (DVs * VP) / 4; i += 256)
    async_g2l_b128((unsigned)(i * 16), src, (unsigned)(i * 16));
  wait_async0();
  __syncthreads();

  int wave = tid >> 5, lane = tid & 31;
  int half = lane >> 4;
  int cbase = lane & 15;

  for (int rt = wave; rt < 16; rt += 8) {  // 16 row-tiles, both N tiles each
    int m0 = rt * 16;
    __builtin_prefetch(&Wo[(m0 + cbase) * DVs], 0, 1);
    v8f acc0 = {0.f, 0.f, 0.f, 0.f, 0.f, 0.f, 0.f, 0.f};
    v8f acc1 = {0.f, 0.f, 0.f, 0.f, 0.f, 0.f, 0.f, 0.f};
    for (int k0 = 0; k0 < DVs; k0 += 4) {
      v2f a = ld_rowk(Wo, DVs, m0, k0, lane);
      v2f b0 = ld_kcol(a_lds, VP, k0, 0, lane);
      v2f b1 = ld_kcol(a_lds, VP, k0, 16, lane);
      acc0 = wmma4(a, b0, acc0);
      acc1 = wmma4(a, b1, acc1);
    }
#pragma unroll
    for (int r = 0; r < 8; ++r) {
      int o = m0 + half * 8 + r;
      int col0 = cbase;
      int col1 = 16 + cbase;
      size_t i0 = ((size_t)(n * Cs + o) * Ts + t) * Vs + col0;
      ybuf[i0] = acc0[r] + bo[o] + x[i0];          // col0 always < 25
      if (col1 < Vs) {
        size_t i1 = ((size_t)(n * Cs + o) * Ts + t) * Vs + col1;
        ybuf[i1] = acc1[r] + bo[o] + x[i1];
      }
    }
  }
}

// ---------------- Kernel E: bn2 stats -> fused scale/shift ----------------
__global__ void kE_bn2_stats(const float* __restrict__ ybuf,
                             const float* __restrict__ gamma,
                             const float* __restrict__ beta,
                             float* __restrict__ bn2) {
  __shared__ float s_sum[256];
  __shared__ float s_sq[256];
  int c = blockIdx.x;
  int tid = threadIdx.x;
  float sum = 0.f, sq = 0.f;
  const int per = Ts * Vs;  // 3200
  for (int i = tid; i < Ns * per; i += 256) {
    int n = i / per, r = i % per;
    float val = ybuf[(size_t)(n * Cs + c) * per + r];
    sum += val;
    sq += val * val;
  }
  s_sum[tid] = sum;
  s_sq[tid] = sq;
  __syncthreads();
  for (int s = 128; s > 0; s >>= 1) {
    if (tid < s) {
      s_sum[tid] += s_sum[tid + s];
      s_sq[tid] += s_sq[tid + s];
    }
    __syncthreads();
  }
  if (tid == 0) {
    const float inv = 1.0f / (float)(Ns * per);
    float mu = s_sum[0] * inv;
    float var = s_sq[0] * inv - mu * mu;
    float sc = rsqrtf(var + EPSc) * gamma[c];
    bn2[c] = sc;
    bn2[Cs + c] = beta[c] - mu * sc;
  }
}

// ---------------- Kernel F: bn2 apply + ReLU ----------------
__global__ void kF_bn2_relu(const float* __restrict__ ybuf,
                            const float* __restrict__ bn2,
                            float* __restrict__ out, size_t total) {
  size_t idx = (size_t)blockIdx.x * blockDim.x + threadIdx.x;
  if (idx >= total) return;
  int c = (int)((idx / (Ts * Vs)) % Cs);
  float v = ybuf[idx] * bn2[c] + bn2[Cs + c];
  out[idx] = v > 0.f ? v : 0.f;
}

// ---------------- launch ----------------
extern "C" void kernel_launch(void* const* d_in, const int* in_sizes, int n_in,
                              void* d_out, int out_size, void* d_ws,
                              size_t ws_size, hipStream_t stream) {
  const float* x      = (const float*)d_in[0];
  const float* qkv_w  = (const float*)d_in[1];
  const float* qkv_b  = (const float*)d_in[2];
  const float* attn_w = (const float*)d_in[3];
  const float* attn_b = (const float*)d_in[4];
  const float* dbn_g  = (const float*)d_in[5];
  const float* dbn_b  = (const float*)d_in[6];
  const float* bn_g   = (const float*)d_in[7];
  const float* bn_b   = (const float*)d_in[8];
  float* out = (float*)d_out;
  float* ws = (float*)d_ws;

  float* dbn     = ws + OFF_DBN;
  float* qkvbuf  = ws + OFF_QKV;
  float* pqkbuf  = ws + OFF_PQK;
  float* attnbuf = ws + OFF_ATT;
  float* bn2     = ws + OFF_BN2;
  float* ybuf    = ws + OFF_QKV;  // alias: qkvbuf dead after kernel C

  (void)hipFuncSetAttribute((const void*)kB_qkv,
                            hipFuncAttributeMaxDynamicSharedMemorySize,
                            2 * Cs * VP * (int)sizeof(float));
  (void)hipFuncSetAttribute((const void*)kC_attn,
                            hipFuncAttributeMaxDynamicSharedMemorySize,
                            NHs * HEADSZ * (int)sizeof(float));
  (void)hipFuncSetAttribute((const void*)kD_proj,
                            hipFuncAttributeMaxDynamicSharedMemorySize,
                            DVs * VP * (int)sizeof(float));

  kA_dbn_stats<<<Cs * Vs, 256, 0, stream>>>(x, dbn_g, dbn_b, dbn);
  kB_qkv<<<Ns * Ts, 256, 2 * Cs * VP * sizeof(float), stream>>>(
      x, qkv_w, qkv_b, dbn, qkvbuf, pqkbuf);
  kC_attn<<<Ns * Ts, 256, NHs * HEADSZ * sizeof(float), stream>>>(
      qkvbuf, pqkbuf, attnbuf);
  kD_proj<<<Ns * Ts, 256, DVs * VP * sizeof(float), stream>>>(
      x, attnbuf, attn_w, attn_b, ybuf);
  kE_bn2_stats<<<Cs, 256, 0, stream>>>(ybuf, bn_g, bn_b, bn2);
  size_t total = (size_t)Ns * Cs * Ts * Vs;
  kF_bn2_relu<<<(unsigned)((total + 255) / 256), 256, 0, stream>>>(ybuf, bn2,
                                                                   out, total);
}